// PathAuxiliarySampler_24275155157096
// MI455X (gfx1250) — compile-verified
//
#include <hip/hip_runtime.h>
#include <math.h>

#define DIM   16384
#define BSIZE 32
#define STEPS 9

typedef float v2f __attribute__((ext_vector_type(2)));
typedef float v8f __attribute__((ext_vector_type(8)));

// ---------------------------------------------------------------------------
// Kernel 1: s = x @ W   (M=32, N=K=16384, fp32)  via V_WMMA_F32_16X16X4_F32.
// Block = 256 threads = 8 waves; wave w owns 16 N-columns, both M-halves.
// Block covers 128 columns; grid = 16384/128 = 128 blocks. W is read exactly
// once, streamed with non-temporal loads (read-once data must not displace
// x/s in the 192MB L2). x chunk staged in LDS (pitch 260: bank-conflict-free
// column broadcast, 8B-aligned pairs -> ds_load_2addr_b64).
// unroll 8 -> 16 NT loads in flight per wave per clause for HBM latency
// hiding at the 23.3 TB/s roofline (1.07 GB of W ~ 46 us floor).
// ---------------------------------------------------------------------------
#define KC 256
__global__ __launch_bounds__(256) void gemm_xw_kernel(
    const float* __restrict__ x, const float* __restrict__ W,
    float* __restrict__ s) {
  __shared__ float xs[32][260];
  const int tid  = threadIdx.x;
  const int lane = tid & 31;
  const int wave = tid >> 5;
  const int m    = lane & 15;   // row within M-half (A) / column N (B, C/D)
  const int half = lane >> 4;   // lane half selects K sub-pair / C row group
  const int n    = blockIdx.x * 128 + wave * 16 + m;

  v8f c0 = {};  // M rows 0..15
  v8f c1 = {};  // M rows 16..31

  for (int k0 = 0; k0 < DIM; k0 += KC) {
    // Cooperative, coalesced load of x[0:32][k0:k0+KC] into LDS.
    for (int i = tid; i < 32 * KC; i += 256) {
      const int r = i >> 8;        // i / KC
      const int c = i & (KC - 1);  // i % KC
      xs[r][c] = x[r * DIM + k0 + c];
    }
    __syncthreads();

#pragma unroll 8
    for (int kk = 0; kk < KC; kk += 4) {
      const int ka = kk + half * 2;  // K index for this lane's VGPR pair
      v2f a0, a1, bm;
      a0.x = xs[m][ka];        a0.y = xs[m][ka + 1];
      a1.x = xs[m + 16][ka];   a1.y = xs[m + 16][ka + 1];
      const long kg = (long)(k0 + ka);
      bm.x = __builtin_nontemporal_load(&W[kg * DIM + n]);
      bm.y = __builtin_nontemporal_load(&W[(kg + 1) * DIM + n]);
      c0 = __builtin_amdgcn_wmma_f32_16x16x4_f32(false, a0, false, bm,
                                                 (short)0, c0, false, false);
      c1 = __builtin_amdgcn_wmma_f32_16x16x4_f32(false, a1, false, bm,
                                                 (short)0, c1, false, false);
    }
    __syncthreads();
  }

  // C/D layout: VGPR v, lanes<16 -> M=v, lanes>=16 -> M=v+8; N = lane%16.
#pragma unroll
  for (int v = 0; v < 8; ++v) {
    s[(v + 8 * half) * DIM + n]        = c0[v];
    s[(16 + v + 8 * half) * DIM + n]   = c1[v];
  }
}

// ---------------------------------------------------------------------------
// Kernel 2: per-row logsumexp of 0.5*(1-2x)*(s+b).  One block per batch row.
// ---------------------------------------------------------------------------
__global__ __launch_bounds__(256) void lse_kernel(
    const float* __restrict__ xv, const float* __restrict__ s,
    const float* __restrict__ bias, float* __restrict__ Z) {
  const int b = blockIdx.x, tid = threadIdx.x;
  __shared__ float red[256];
  const float* xrow = xv + (size_t)b * DIM;
  const float* srow = s + (size_t)b * DIM;

  float mx = -__builtin_inff();
  for (int j = tid; j < DIM; j += 256) {
    const float v = (1.f - 2.f * xrow[j]) * (srow[j] + bias[j]) * 0.5f;
    mx = fmaxf(mx, v);
  }
  red[tid] = mx;
  __syncthreads();
  for (int off = 128; off > 0; off >>= 1) {
    if (tid < off) red[tid] = fmaxf(red[tid], red[tid + off]);
    __syncthreads();
  }
  mx = red[0];
  __syncthreads();

  float sm = 0.f;
  for (int j = tid; j < DIM; j += 256) {
    const float v = (1.f - 2.f * xrow[j]) * (srow[j] + bias[j]) * 0.5f;
    sm += __expf(v - mx);
  }
  red[tid] = sm;
  __syncthreads();
  for (int off = 128; off > 0; off >>= 1) {
    if (tid < off) red[tid] += red[tid + off];
    __syncthreads();
  }
  if (tid == 0) Z[b] = mx + __logf(red[0]);
}

// ---------------------------------------------------------------------------
// Kernel 3: the 9 Gumbel-max steps. Rows are independent -> one block per row
// runs its whole chain. Each step: argmax(0.5*(1-2y)*(s+b) + g), flip one bit,
// rank-1 update s += sign * W[idx,:]. Steps with t >= radius are exact no-ops
// (the mask never flips back on), so we terminate early.
// ---------------------------------------------------------------------------
__global__ __launch_bounds__(256) void steps_kernel(
    const float* __restrict__ x, const float* __restrict__ W,
    const float* __restrict__ bias, const float* __restrict__ gumbel,
    const int* __restrict__ radius, float* __restrict__ s,
    float* __restrict__ y) {
  const int b = blockIdx.x, tid = threadIdx.x;
  __shared__ float sVal[256];
  __shared__ int   sIdx[256];
  __shared__ int   flipIdx;
  __shared__ float flipSgn;

  float*       yrow = y + (size_t)b * DIM;
  float*       srow = s + (size_t)b * DIM;
  const float* xrow = x + (size_t)b * DIM;

  for (int j = tid; j < DIM; j += 256) yrow[j] = xrow[j];
  __syncthreads();

  const int r = radius[b];                 // in [1, 9]
  const int nsteps = (r < STEPS) ? r : STEPS;

  for (int t = 0; t < nsteps; ++t) {
    const float* grow = gumbel + ((size_t)t * BSIZE + b) * DIM;

    float best = -__builtin_inff();
    int bestIdx = 0;
    for (int j = tid; j < DIM; j += 256) {
      const float v =
          (1.f - 2.f * yrow[j]) * (srow[j] + bias[j]) * 0.5f + grow[j];
      if (v > best) { best = v; bestIdx = j; }   // keep first (lowest j)
    }
    sVal[tid] = best; sIdx[tid] = bestIdx;
    __syncthreads();
    for (int off = 128; off > 0; off >>= 1) {
      if (tid < off) {
        const float ov = sVal[tid + off];
        const int   oi = sIdx[tid + off];
        if (ov > sVal[tid] || (ov == sVal[tid] && oi < sIdx[tid])) {
          sVal[tid] = ov; sIdx[tid] = oi;
        }
      }
      __syncthreads();
    }
    if (tid == 0) {
      const int   idx = sIdx[0];
      const float cur = yrow[idx];
      flipIdx = idx;
      flipSgn = 1.f - 2.f * cur;   // delta applied to x_idx by the flip
      yrow[idx] = 1.f - cur;
    }
    __syncthreads();
    const int   idx = flipIdx;
    const float sg  = flipSgn;
    const float* wrow = W + (size_t)idx * DIM;
    for (int j = tid; j < DIM; j += 256) srow[j] += sg * wrow[j];
    __syncthreads();
  }
}

// ---------------------------------------------------------------------------
// Kernel 4: accept test + output write.
// ---------------------------------------------------------------------------
__global__ __launch_bounds__(256) void accept_kernel(
    const float* __restrict__ x, const float* __restrict__ y,
    const float* __restrict__ Zx, const float* __restrict__ Zy,
    const float* __restrict__ accept_u, float* __restrict__ out) {
  const int b = blockIdx.x, tid = threadIdx.x;
  const float acc = (__expf(Zx[b] - Zy[b]) >= accept_u[b]) ? 1.f : 0.f;
  const float* xrow = x + (size_t)b * DIM;
  const float* yrow = y + (size_t)b * DIM;
  float*       orow = out + (size_t)b * DIM;
  for (int j = tid; j < DIM; j += 256)
    orow[j] = acc * yrow[j] + (1.f - acc) * xrow[j];
}

// ---------------------------------------------------------------------------
extern "C" void kernel_launch(void* const* d_in, const int* in_sizes, int n_in,
                              void* d_out, int out_size, void* d_ws,
                              size_t ws_size, hipStream_t stream) {
  const float* x        = (const float*)d_in[0];
  const float* W        = (const float*)d_in[1];
  const float* bias     = (const float*)d_in[2];
  const float* gumbel   = (const float*)d_in[3];
  const float* accept_u = (const float*)d_in[4];
  const int*   radius   = (const int*)d_in[5];
  float*       out      = (float*)d_out;

  char* ws = (char*)d_ws;
  float* s  = (float*)ws;                                     // 32*16384 f32
  float* y  = (float*)(ws + (size_t)BSIZE * DIM * 4);         // 32*16384 f32
  float* Zx = (float*)(ws + 2 * (size_t)BSIZE * DIM * 4);     // 32 f32
  float* Zy = Zx + BSIZE;                                     // 32 f32

  gemm_xw_kernel<<<DIM / 128, 256, 0, stream>>>(x, W, s);
  lse_kernel<<<BSIZE, 256, 0, stream>>>(x, s, bias, Zx);
  steps_kernel<<<BSIZE, 256, 0, stream>>>(x, W, bias, gumbel, radius, s, y);
  lse_kernel<<<BSIZE, 256, 0, stream>>>(y, s, bias, Zy);
  accept_kernel<<<BSIZE, 256, 0, stream>>>(x, y, Zx, Zy, accept_u, out);
}